// SimpleGATConv_1056561955466
// MI455X (gfx1250) — compile-verified
//
#include <hip/hip_runtime.h>
#include <hip/hip_bf16.h>

#define NN 50000
#define EE 1600000
#define INF 256
#define HH 4
#define DD 64
#define HD 256          // H*D
#define NEG_SLOPE 0.2f

typedef __attribute__((ext_vector_type(2))) float v2f;
typedef __attribute__((ext_vector_type(8))) float v8f;

// ---- monotone float<->uint mapping for atomic max on floats ----
__device__ __forceinline__ unsigned flipf(float f) {
    unsigned u = __float_as_uint(f);
    return (u & 0x80000000u) ? ~u : (u | 0x80000000u);
}
__device__ __forceinline__ float unflipf(unsigned u) {
    return __uint_as_float((u & 0x80000000u) ? (u & 0x7FFFFFFFu) : ~u);
}
__device__ __forceinline__ float lrelu(float v) {
    return v > 0.0f ? v : NEG_SLOPE * v;
}

// ---------------------------------------------------------------------------
// K0: init out = bias (broadcast), m = 0 (flipped -inf floor), s = 0
// ---------------------------------------------------------------------------
__global__ __launch_bounds__(256)
void gat_init_kernel(float* __restrict__ out, const float* __restrict__ bias,
                     unsigned* __restrict__ m_u, float* __restrict__ ssum) {
    const int i = blockIdx.x * 256 + threadIdx.x;
    if (i < NN * HD) out[i] = bias[i & (HD - 1)];
    if (i < NN * HH) { m_u[i] = 0u; ssum[i] = 0.0f; }
}

// ---------------------------------------------------------------------------
// K1: feat = x @ W via V_WMMA_F32_16X16X4_F32, fused el/er epilogue.
// Block = 512 threads (16 waves). Block handles 16 rows x 256 cols.
// Wave w computes the 16x16 tile at columns [16w, 16w+16).
// ---------------------------------------------------------------------------
__global__ __launch_bounds__(512)
void gat_gemm_kernel(const float* __restrict__ x, const float* __restrict__ W,
                     const float* __restrict__ attn_l, const float* __restrict__ attn_r,
                     float* __restrict__ feat, float* __restrict__ el,
                     float* __restrict__ er) {
    __shared__ float lds_x[16 * 260];         // padded stride: conflict-free A reads
    __shared__ float lds_el[16 * HH];
    __shared__ float lds_er[16 * HH];

    const int tid  = threadIdx.x;
    const int row0 = blockIdx.x * 16;

    // Coalesced stage of the 16x256 x-tile into LDS (8 floats/thread)
    for (int i = tid; i < 16 * 256; i += 512) {
        const int r = i >> 8, c = i & 255;
        lds_x[r * 260 + c] = x[(row0 + r) * INF + c];
    }
    if (tid < 16 * HH) { lds_el[tid] = 0.0f; lds_er[tid] = 0.0f; }
    __syncthreads();

    const int wave = tid >> 5;        // 0..15 -> column tile
    const int lane = tid & 31;
    const int col0 = wave * 16;
    const int mrow = lane & 15;       // A: M index
    const int half = lane >> 4;       // selects K pair {0,1} vs {2,3}
    const int ncol = lane & 15;       // B/C: N index

    v8f c = {};
    for (int k0 = 0; k0 < INF; k0 += 4) {
        const int k = k0 + 2 * half;
        v2f a, b;
        a.x = lds_x[mrow * 260 + k];
        a.y = lds_x[mrow * 260 + k + 1];
        b.x = W[(k)     * HD + col0 + ncol];
        b.y = W[(k + 1) * HD + col0 + ncol];
        c = __builtin_amdgcn_wmma_f32_16x16x4_f32(
                /*neg_a=*/false, a, /*neg_b=*/false, b,
                /*c_mod=*/(short)0, c, /*reuse_a=*/false, /*reuse_b=*/false);
    }

    // Per-lane attention coefficients for this column
    const int col = col0 + ncol;
    const int h   = col >> 6;         // head (uniform within a 16-col tile)
    const int d   = col & 63;
    const float al = attn_l[h * DD + d];
    const float ar = attn_r[h * DD + d];

    // C layout: VGPR r holds element (m = r + 8*half, n = lane&15)
#pragma unroll
    for (int r = 0; r < 8; ++r) {
        const int   mr = r + 8 * half;
        const float v  = c[r];
        feat[(row0 + mr) * HD + col] = v;
        atomicAdd(&lds_el[mr * HH + h], v * al);   // ds_add_f32
        atomicAdd(&lds_er[mr * HH + h], v * ar);
    }
    __syncthreads();
    if (tid < 16 * HH) {
        const int mr = tid >> 2, hh = tid & 3;
        el[(row0 + mr) * HH + hh] = lds_el[tid];
        er[(row0 + mr) * HH + hh] = lds_er[tid];
    }
}

// ---------------------------------------------------------------------------
// K2: per-edge leaky-relu score, segment max over dst via u32 atomicMax
// ---------------------------------------------------------------------------
__global__ __launch_bounds__(256)
void gat_edge_max_kernel(const int* __restrict__ src, const int* __restrict__ dst,
                         const float* __restrict__ el, const float* __restrict__ er,
                         unsigned* __restrict__ m_u) {
    const int e = blockIdx.x * 256 + threadIdx.x;
    if (e >= EE) return;
    const int s = src[e], d = dst[e];
#pragma unroll
    for (int h = 0; h < HH; ++h) {
        const float v = lrelu(el[s * HH + h] + er[d * HH + h]);
        atomicMax(&m_u[d * HH + h], flipf(v));
    }
}

// ---------------------------------------------------------------------------
// K3: segment sum of exp(e - m[dst]) via f32 atomic add
// ---------------------------------------------------------------------------
__global__ __launch_bounds__(256)
void gat_edge_sum_kernel(const int* __restrict__ src, const int* __restrict__ dst,
                         const float* __restrict__ el, const float* __restrict__ er,
                         const unsigned* __restrict__ m_u, float* __restrict__ ssum) {
    const int e = blockIdx.x * 256 + threadIdx.x;
    if (e >= EE) return;
    const int s = src[e], d = dst[e];
#pragma unroll
    for (int h = 0; h < HH; ++h) {
        const float v  = lrelu(el[s * HH + h] + er[d * HH + h]);
        const float ex = __expf(v - unflipf(m_u[d * HH + h]));
        unsafeAtomicAdd(&ssum[d * HH + h], ex);    // global_atomic_add_f32
    }
}

// ---------------------------------------------------------------------------
// K4: one wave per edge: out[dst] += alpha * feat[src]
//   lanes 0..3 compute the 4 per-head alphas, broadcast via wave32 shfl;
//   each lane gathers 2 float4 of feat (coalesced b128) and does 8 f32 atomics.
// ---------------------------------------------------------------------------
__global__ __launch_bounds__(256)
void gat_edge_aggr_kernel(const int* __restrict__ src, const int* __restrict__ dst,
                          const float* __restrict__ el, const float* __restrict__ er,
                          const unsigned* __restrict__ m_u, const float* __restrict__ ssum,
                          const float* __restrict__ feat, float* __restrict__ out) {
    const int lane = threadIdx.x & 31;
    const int e    = blockIdx.x * 8 + (threadIdx.x >> 5);
    if (e >= EE) return;
    const int s = src[e], d = dst[e];

    float alpha = 0.0f;
    if (lane < HH) {
        const float v  = lrelu(el[s * HH + lane] + er[d * HH + lane]);
        const float mm = unflipf(m_u[d * HH + lane]);
        alpha = __expf(v - mm) / ssum[d * HH + lane];
    }
    const float a0 = __shfl(alpha, 0, 32);
    const float a1 = __shfl(alpha, 1, 32);
    const float a2 = __shfl(alpha, 2, 32);
    const float a3 = __shfl(alpha, 3, 32);

    // lane covers cols [4*lane, 4*lane+3] and [128 + 4*lane, 128 + 4*lane+3];
    // each float4 lies entirely inside one head (64-col heads)
    const float4 f0 = *(const float4*)(feat + s * HD + lane * 4);
    const float4 f1 = *(const float4*)(feat + s * HD + 128 + lane * 4);
    const float aa0 = (lane >> 4) ? a1 : a0;   // head 0 or 1
    const float aa1 = (lane >> 4) ? a3 : a2;   // head 2 or 3

    float* o0 = out + d * HD + lane * 4;
    unsafeAtomicAdd(o0 + 0, f0.x * aa0);
    unsafeAtomicAdd(o0 + 1, f0.y * aa0);
    unsafeAtomicAdd(o0 + 2, f0.z * aa0);
    unsafeAtomicAdd(o0 + 3, f0.w * aa0);
    float* o1 = out + d * HD + 128 + lane * 4;
    unsafeAtomicAdd(o1 + 0, f1.x * aa1);
    unsafeAtomicAdd(o1 + 1, f1.y * aa1);
    unsafeAtomicAdd(o1 + 2, f1.z * aa1);
    unsafeAtomicAdd(o1 + 3, f1.w * aa1);
}

// ---------------------------------------------------------------------------
extern "C" void kernel_launch(void* const* d_in, const int* in_sizes, int n_in,
                              void* d_out, int out_size, void* d_ws, size_t ws_size,
                              hipStream_t stream) {
    const float* x      = (const float*)d_in[0];
    const int*   src    = (const int*)d_in[1];
    const int*   dst    = (const int*)d_in[2];
    const float* W      = (const float*)d_in[3];
    const float* attn_l = (const float*)d_in[4];
    const float* attn_r = (const float*)d_in[5];
    const float* bias   = (const float*)d_in[6];
    float* out = (float*)d_out;

    char* ws = (char*)d_ws;
    float*    feat = (float*)ws;                                   // 51.2 MB
    float*    el   = (float*)(ws + (size_t)NN * HD * sizeof(float));
    float*    er   = el + (size_t)NN * HH;                         // 0.8 MB each
    unsigned* m_u  = (unsigned*)(er + (size_t)NN * HH);
    float*    ssum = (float*)(m_u + (size_t)NN * HH);

    // init out=bias, m=floor, s=0   (covers NN*HD = 12.8M elements)
    gat_init_kernel<<<(NN * HD + 255) / 256, 256, 0, stream>>>(out, bias, m_u, ssum);

    // projection + el/er  (50000 = 3125 * 16 rows exactly)
    gat_gemm_kernel<<<NN / 16, 512, 0, stream>>>(x, W, attn_l, attn_r, feat, el, er);

    // segment max / segment sum over edges (1.6M = 6250 * 256 exactly)
    gat_edge_max_kernel<<<EE / 256, 256, 0, stream>>>(src, dst, el, er, m_u);
    gat_edge_sum_kernel<<<EE / 256, 256, 0, stream>>>(src, dst, el, er, m_u, ssum);

    // weighted scatter-aggregate: one wave per edge (1.6M = 200000 * 8)
    gat_edge_aggr_kernel<<<EE / 8, 256, 0, stream>>>(src, dst, el, er, m_u, ssum,
                                                     feat, out);
}